// SoftEmbedding_42786464202989
// MI455X (gfx1250) — compile-verified
//
#include <hip/hip_runtime.h>
#include <hip/hip_bf16.h>

// Problem constants (match reference)
static constexpr int V = 50257;
static constexpr int D = 1024;
static constexpr int B = 16;
static constexpr int S = 2048;
static constexpr int P = 10;
static constexpr int A = 5;
static constexpr int T = 64;
static constexpr int L = P + S + A + T;   // 2127 = 3 * 709
static constexpr int ROWS_PER_BLOCK = 3;  // divides L exactly
static constexpr int BLOCKS_PER_BATCH = L / ROWS_PER_BLOCK; // 709

// ---------------------------------------------------------------------------
// Kernel 1: per-batch split computation.
// split = argmax(1 - concat(ones(P), mask)) + 1.  Since the prepend part is
// all ones, the first zero (if any) lies in the mask region; if the mask is
// all ones, argmax returns 0 -> split = 1 (reference semantics).
// ws[b] = split2 = split + A.
// ---------------------------------------------------------------------------
__global__ void k_split(const int* __restrict__ mask, int* __restrict__ ws) {
    const int b   = blockIdx.x;
    const int tid = threadIdx.x;
    __shared__ int red[256];

    int mn = S; // sentinel: "no zero found"
    for (int j = tid; j < S; j += 256) {
        if (mask[b * S + j] == 0) { mn = j; break; } // j increases -> first hit is thread-local min
    }
    red[tid] = mn;
    __syncthreads();
    for (int s = 128; s > 0; s >>= 1) {
        if (tid < s) {
            int o = red[tid + s];
            if (o < red[tid]) red[tid] = o;
        }
        __syncthreads();
    }
    if (tid == 0) {
        const int fz    = red[0];
        const int split = (fz < S) ? (P + fz + 1) : 1;
        ws[b] = split + A; // split2
    }
}

// ---------------------------------------------------------------------------
// Kernel 2: emb assembly via the CDNA5 async data-mover path.
// Each block copies ROWS_PER_BLOCK rows of 4KB. Per row, each lane moves 16B:
//   global_load_async_to_lds_b128  (memory -> LDS, ASYNCcnt, RT so repeated
//                                   wte rows can hit L2)
//   s_wait_asynccnt 0
//   global_store_async_from_lds_b128 th:TH_STORE_NT scope:SCOPE_DEV
//                                  (LDS -> memory, non-temporal: the 139MB
//                                   output is write-once, keep L2 for wte)
// Double-buffered LDS lets row r's store overlap row r+1's gather.
// Each wave loads/stores its own disjoint 512B LDS chunk -> no barriers.
// ---------------------------------------------------------------------------
__global__ void __launch_bounds__(256) k_emb(
    const int*   __restrict__ ids,    // (B, S)
    const int*   __restrict__ tids,   // (B, T)
    const float* __restrict__ wte,    // (V, D)
    const float* __restrict__ pre,    // (P, D)
    const float* __restrict__ app,    // (A, D)
    const int*   __restrict__ ws,     // (B,) split2
    float*       __restrict__ out_emb)// (B, L, D)
{
    const int b   = blockIdx.x / BLOCKS_PER_BATCH;
    const int j0  = (blockIdx.x % BLOCKS_PER_BATCH) * ROWS_PER_BLOCK;
    const int tid = threadIdx.x;

    const int sp2 = ws[b];
    const int sp  = sp2 - A;

    __shared__ __align__(16) float lds[2][D];
    // Generic pointer low 32 bits == LDS byte address (LDS aperture mapping).
    const unsigned lds_off0 = (unsigned)(unsigned long long)(&lds[0][0]) + (unsigned)tid * 16u;
    const unsigned lds_off1 = (unsigned)(unsigned long long)(&lds[1][0]) + (unsigned)tid * 16u;

    // Final layout: [emb0[0:sp], append[0:A], target_emb[0:T], emb0[sp:P+S]]
    auto srcptr = [&](int j) -> const float* {
        if (j < sp)      return (j < P) ? pre + (size_t)j * D
                                        : wte + (size_t)ids[b * S + (j - P)] * D;
        if (j < sp2)     return app + (size_t)(j - sp) * D;
        if (j < sp2 + T) return wte + (size_t)tids[b * T + (j - sp2)] * D;
        const int jp = j - (A + T); // index into emb0; jp >= sp (sp = 1 < P if mask has no zero)
        return (jp < P) ? pre + (size_t)jp * D
                        : wte + (size_t)ids[b * S + (jp - P)] * D;
    };

    // Prologue: kick off gather of row 0 into buffer 0.
    {
        const unsigned long long ga = (unsigned long long)(srcptr(j0) + tid * 4);
        asm volatile("global_load_async_to_lds_b128 %0, %1, off"
                     :: "v"(lds_off0), "v"(ga) : "memory");
    }

#pragma unroll
    for (int r = 0; r < ROWS_PER_BLOCK; ++r) {
        const int j = j0 + r;
        const unsigned loff = (r & 1) ? lds_off1 : lds_off0;

        // Ensure load of row r (and store of row r-1) completed.
        asm volatile("s_wait_asynccnt 0" ::: "memory");

        const unsigned long long gout =
            (unsigned long long)(out_emb + ((size_t)b * L + j) * D + tid * 4);
        asm volatile("global_store_async_from_lds_b128 %0, %1, off th:TH_STORE_NT scope:SCOPE_DEV"
                     :: "v"(gout), "v"(loff) : "memory");

        if (r + 1 < ROWS_PER_BLOCK) {
            const unsigned lnext = ((r + 1) & 1) ? lds_off1 : lds_off0;
            const unsigned long long ga =
                (unsigned long long)(srcptr(j + 1) + tid * 4);
            asm volatile("global_load_async_to_lds_b128 %0, %1, off"
                         :: "v"(lnext), "v"(ga) : "memory");
        }
    }
    // S_ENDPGM performs an implicit wait-idle, covering the final async store.
}

// ---------------------------------------------------------------------------
// Kernel 3: am, target_mask (shifted right by 1), labels (shifted left by 1),
// target_ids (shifted left by 1), split2. All written as float (d_out is a
// float buffer; integer outputs are value-converted).
// ---------------------------------------------------------------------------
__global__ void k_small(const int* __restrict__ mask,   // (B, S)
                        const int* __restrict__ tids,   // (B, T)
                        const int* __restrict__ tam,    // (B, T)
                        const int* __restrict__ ws,     // (B,)
                        float* __restrict__ out_am,     // (B, L)
                        float* __restrict__ out_tm,     // (B, L)
                        float* __restrict__ out_tids,   // (B, T)
                        float* __restrict__ out_sp2,    // (B,)
                        float* __restrict__ out_lab)    // (B, L)
{
    const int b   = blockIdx.x;
    const int tid = threadIdx.x;
    const int sp2 = ws[b];
    const int sp  = sp2 - A;

    auto raw_tm = [&](int q) -> float {
        return (q >= sp2 && q < sp2 + T) ? (float)tam[b * T + (q - sp2)] : 0.0f;
    };

    for (int j = tid; j < L; j += 256) {
        // am: [am0[0:sp], ones(A), tam[0:T], am0[sp:P+S]]
        float amv;
        if (j < sp)           amv = (j < P) ? 1.0f : (float)mask[b * S + (j - P)];
        else if (j < sp2)     amv = 1.0f;
        else if (j < sp2 + T) amv = (float)tam[b * T + (j - sp2)];
        else {
            const int jp = j - (A + T);
            amv = (jp < P) ? 1.0f : (float)mask[b * S + (jp - P)];
        }
        out_am[b * L + j] = amv;

        // target_mask shifted right by one with leading zero
        out_tm[b * L + j] = (j == 0) ? 0.0f : raw_tm(j - 1);

        // labels: raw labels shifted left by one with trailing -1
        int lab = -1;
        if (j < L - 1) {
            const int q = j + 1;
            if (raw_tm(q) == 1.0f) {
                int idx = q - sp2;
                idx = idx < 0 ? 0 : (idx > T - 1 ? T - 1 : idx);
                lab = tids[b * T + idx];
            }
        }
        out_lab[b * L + j] = (float)lab;
    }

    if (tid < T)
        out_tids[b * T + tid] = (float)((tid < T - 1) ? tids[b * T + tid + 1] : -1);
    if (tid == 0)
        out_sp2[b] = (float)sp2;
}

// ---------------------------------------------------------------------------
// Launch: outputs concatenated flat in return order:
//   emb (B*L*D), am (B*L), target_mask (B*L), target_ids (B*T), split2 (B),
//   labels (B*L)
// ---------------------------------------------------------------------------
extern "C" void kernel_launch(void* const* d_in, const int* in_sizes, int n_in,
                              void* d_out, int out_size, void* d_ws, size_t ws_size,
                              hipStream_t stream) {
    const int*   input_ids = (const int*)d_in[0];
    const int*   attn      = (const int*)d_in[1];
    const int*   tgt_ids   = (const int*)d_in[2];
    const int*   tam       = (const int*)d_in[3];
    const float* wte       = (const float*)d_in[4];
    const float* pre       = (const float*)d_in[5];
    const float* app       = (const float*)d_in[6];

    float* out      = (float*)d_out;
    float* out_emb  = out;
    float* out_am   = out_emb + (size_t)B * L * D;
    float* out_tm   = out_am  + (size_t)B * L;
    float* out_tids = out_tm  + (size_t)B * L;
    float* out_sp2  = out_tids + (size_t)B * T;
    float* out_lab  = out_sp2 + (size_t)B;

    int* ws = (int*)d_ws;

    k_split<<<B, 256, 0, stream>>>(attn, ws);
    k_emb<<<B * BLOCKS_PER_BATCH, 256, 0, stream>>>(
        input_ids, tgt_ids, wte, pre, app, ws, out_emb);
    k_small<<<B, 256, 0, stream>>>(attn, tgt_ids, tam, ws,
                                   out_am, out_tm, out_tids, out_sp2, out_lab);
}